// MicroDLRMDRAM_82497731822232
// MI455X (gfx1250) — compile-verified
//
#include <hip/hip_runtime.h>

typedef __attribute__((ext_vector_type(2))) float v2f;
typedef __attribute__((ext_vector_type(8))) float v8f;

#define BDIM 256
#define ROWS 128           // batch rows per block (8 wave32 x 16-row WMMA tiles)
#define NWAVE 8
#define ZS   204           // zbuf row stride (floats): 16B-aligned rows, conflict-free
#define W0TS 204
#define W1TS 34
#define H1S  34
#define H2S  18
#define VTAB 2000000ull

__global__ __launch_bounds__(BDIM) void dlrm_fused(
    const float* __restrict__ dense_x,
    const long long* __restrict__ sparse_idx,
    const long long* __restrict__ hash_seeds,
    const float* __restrict__ emb_table,
    const float* __restrict__ bot_w0, const float* __restrict__ bot_b0,
    const float* __restrict__ bot_w1, const float* __restrict__ bot_b1,
    const float* __restrict__ top_w0, const float* __restrict__ top_b0,
    const float* __restrict__ top_w1, const float* __restrict__ top_b1,
    const float* __restrict__ top_w2, const float* __restrict__ top_b2,
    float* __restrict__ out, int batch)
{
    __shared__ float zbuf[ROWS * ZS];          // 128 x 200 concat features (padded)
    __shared__ float w0t[32 * W0TS];           // top_w0 transposed [n][k]
    __shared__ float w1t[16 * W1TS];           // top_w1 transposed [n][k]
    __shared__ float h1buf[NWAVE * 16 * H1S];  // per-wave 16x32 hidden
    __shared__ float h2buf[NWAVE * 16 * H2S];  // per-wave 16x16 hidden
    __shared__ float sw[288];                  // small weights/biases
    __shared__ unsigned int hidx[ROWS * 3];    // hashed embedding rows

    const int t = threadIdx.x;
    const int rowBase = blockIdx.x * ROWS;

    // ---- Phase 0: stage weights into LDS ----
    for (int i = t; i < 273; i += BDIM) {
        float v;
        if (i < 128)      v = bot_w0[i];
        else if (i < 136) v = bot_b0[i - 128];
        else if (i < 200) v = bot_w1[i - 136];
        else if (i < 208) v = bot_b1[i - 200];
        else if (i < 240) v = top_b0[i - 208];
        else if (i < 256) v = top_b1[i - 240];
        else if (i < 272) v = top_w2[i - 256];
        else              v = top_b2[0];
        sw[i] = v;
    }
    for (int i = t; i < 200 * 32; i += BDIM) {       // coalesced read of top_w0
        int k = i >> 5, n = i & 31;
        w0t[n * W0TS + k] = top_w0[i];
    }
    for (int i = t; i < 32 * 16; i += BDIM) {
        int k = i >> 4, n = i & 15;
        w1t[n * W1TS + k] = top_w1[i];
    }
    __syncthreads();

    // ---- Phase 1: bot MLP (threads 0-127) + int64 hashes (threads 128-255) ----
    if (t < ROWS) {
        int row = rowBase + t;
        if (row < batch) {
            float d[16];
            #pragma unroll
            for (int i = 0; i < 16; i++) d[i] = dense_x[row * 16 + i];
            float h0[8];
            #pragma unroll
            for (int o = 0; o < 8; o++) {
                float s = sw[128 + o];
                #pragma unroll
                for (int i = 0; i < 16; i++) s += d[i] * sw[i * 8 + o];
                h0[o] = fmaxf(s, 0.0f);
            }
            #pragma unroll
            for (int o = 0; o < 8; o++) {
                float s = sw[200 + o];
                #pragma unroll
                for (int i = 0; i < 8; i++) s += h0[i] * sw[136 + i * 8 + o];
                zbuf[t * ZS + o] = fmaxf(s, 0.0f);
            }
        }
    } else {
        int r = t - ROWS;
        int row = rowBase + r;
        if (row < batch) {
            #pragma unroll
            for (int f = 0; f < 3; f++) {
                // splitmix64-style hash with signed-int64 wraparound semantics
                unsigned long long x =
                    (unsigned long long)sparse_idx[(long long)f * batch + row] ^
                    (unsigned long long)hash_seeds[f];
                x = (x ^ (unsigned long long)((long long)x >> 30)) * 13787848793156543929ull;
                x = (x ^ (unsigned long long)((long long)x >> 27)) * 10723151780598845931ull;
                x = x ^ (unsigned long long)((long long)x >> 31);
                long long s = (long long)x;
                unsigned long long a = (s < 0) ? (0ull - (unsigned long long)s)
                                               : (unsigned long long)s;
                hidx[r * 3 + f] = (unsigned int)(a % VTAB);
            }
        }
    }
    __syncthreads();

    // ---- Phase 2: async gather memory -> LDS (no VGPR staging, ASYNCcnt) ----
    // 16 consecutive lanes stream one contiguous 256B embedding row.
    for (int i = t; i < ROWS * 3 * 16; i += BDIM) {
        int q = i & 15;          // float4 index within row
        int p = i >> 4;          // (feature,row) pair
        int f = p >> 7;          // p / ROWS
        int r = p & (ROWS - 1);
        if (rowBase + r < batch) {
            unsigned long long gaddr = (unsigned long long)(const void*)
                (emb_table + (size_t)hidx[r * 3 + f] * 64 + (size_t)q * 4);
            // LDS byte offset (low 32 bits of generic pointer into __shared__)
            unsigned ldsoff = (unsigned)(size_t)(const void*)
                (zbuf + r * ZS + 8 + f * 64 + q * 4);
            asm volatile("global_load_async_to_lds_b128 %0, %1, off"
                         :: "v"(ldsoff), "v"(gaddr) : "memory");
        }
    }
    asm volatile("s_wait_asynccnt 0x0" ::: "memory");
    __syncthreads();

    // ---- Phase 3: top layer0 GEMM via V_WMMA_F32_16X16X4_F32 ----
    // A fragment: lane holds row m = lane&15; K = 2*(lane>=16) + vgpr
    // D fragment: row M = vgpr + 8*(lane>=16), col N = lane&15
    const int lane = t & 31;
    const int wv = t >> 5;
    const int m = lane & 15;
    const int kh = (lane >> 4) & 1;
    const float* zrow = zbuf + (wv * 16 + m) * ZS;

    v8f acc0 = {0.f,0.f,0.f,0.f,0.f,0.f,0.f,0.f};
    v8f acc1 = {0.f,0.f,0.f,0.f,0.f,0.f,0.f,0.f};
    #pragma unroll 2
    for (int k0 = 0; k0 < 200; k0 += 4) {
        v2f a  = *(const v2f*)(zrow + k0 + 2 * kh);
        v2f b0 = *(const v2f*)(w0t + m * W0TS + k0 + 2 * kh);          // N-tile 0
        v2f b1 = *(const v2f*)(w0t + (16 + m) * W0TS + k0 + 2 * kh);   // N-tile 1
        acc0 = __builtin_amdgcn_wmma_f32_16x16x4_f32(false, a, false, b0,
                                                     (short)0, acc0, false, false);
        acc1 = __builtin_amdgcn_wmma_f32_16x16x4_f32(false, a, false, b1,
                                                     (short)0, acc1, false, false);
    }
    {
        float bias0 = sw[208 + m];
        float bias1 = sw[208 + 16 + m];
        float* h1 = h1buf + wv * 16 * H1S;
        #pragma unroll
        for (int v = 0; v < 8; v++) {
            int rm = v + 8 * kh;
            h1[rm * H1S + m]      = fmaxf(acc0[v] + bias0, 0.0f);
            h1[rm * H1S + 16 + m] = fmaxf(acc1[v] + bias1, 0.0f);
        }
    }
    __syncthreads();

    // ---- Phase 4: top layer1 GEMM (16x32 @ 32x16) ----
    v8f acc2 = {0.f,0.f,0.f,0.f,0.f,0.f,0.f,0.f};
    {
        const float* h1 = h1buf + wv * 16 * H1S;
        #pragma unroll
        for (int k0 = 0; k0 < 32; k0 += 4) {
            v2f a = *(const v2f*)(h1 + m * H1S + k0 + 2 * kh);
            v2f b = *(const v2f*)(w1t + m * W1TS + k0 + 2 * kh);
            acc2 = __builtin_amdgcn_wmma_f32_16x16x4_f32(false, a, false, b,
                                                         (short)0, acc2, false, false);
        }
        float bias = sw[240 + m];
        float* h2 = h2buf + wv * 16 * H2S;
        #pragma unroll
        for (int v = 0; v < 8; v++)
            h2[(v + 8 * kh) * H2S + m] = fmaxf(acc2[v] + bias, 0.0f);
    }
    __syncthreads();

    // ---- Phase 5: layer2 (16 -> 1) + sigmoid ----
    if (t < ROWS) {
        int row = rowBase + t;
        if (row < batch) {
            const float* h2 = h2buf + (t >> 4) * 16 * H2S + (t & 15) * H2S;
            float s = sw[272];
            #pragma unroll
            for (int n = 0; n < 16; n++) s += h2[n] * sw[256 + n];
            out[row] = 1.0f / (1.0f + __expf(-s));
        }
    }
}

extern "C" void kernel_launch(void* const* d_in, const int* in_sizes, int n_in,
                              void* d_out, int out_size, void* d_ws, size_t ws_size,
                              hipStream_t stream) {
    const float*     dense_x    = (const float*)d_in[0];
    const long long* sparse_idx = (const long long*)d_in[1];
    // d_in[2] = sparse_offsets == arange(B): one index per bag -> pure gather
    const long long* hash_seeds = (const long long*)d_in[3];
    const float*     emb_table  = (const float*)d_in[4];
    const float*     bot_w0     = (const float*)d_in[5];
    const float*     bot_b0     = (const float*)d_in[6];
    const float*     bot_w1     = (const float*)d_in[7];
    const float*     bot_b1     = (const float*)d_in[8];
    const float*     top_w0     = (const float*)d_in[9];
    const float*     top_b0     = (const float*)d_in[10];
    const float*     top_w1     = (const float*)d_in[11];
    const float*     top_b1     = (const float*)d_in[12];
    const float*     top_w2     = (const float*)d_in[13];
    const float*     top_b2     = (const float*)d_in[14];

    int batch = in_sizes[0] / 16;              // dense_x is (B, 16)
    int grid = (batch + ROWS - 1) / ROWS;

    dlrm_fused<<<grid, BDIM, 0, stream>>>(
        dense_x, sparse_idx, hash_seeds, emb_table,
        bot_w0, bot_b0, bot_w1, bot_b1,
        top_w0, top_b0, top_w1, top_b1, top_w2, top_b2,
        (float*)d_out, batch);
}